// Router_22428319220045
// MI455X (gfx1250) — compile-verified
//
#include <hip/hip_runtime.h>
#include <hip/hip_bf16.h>

// MoE router (experts-choose) for MI455X / gfx1250.
//
// Roofline: out writes 537 MB + in reads 134 MB ≈ 29 us @ 23.3 TB/s; the
// router GEMM is only 2.15 GFLOP, so everything is memory-bound. Strategy:
//   1) hipMemsetAsync zeros the (almost entirely zero) combine/dispatch
//      outputs and the z-loss slot at pure store bandwidth.
//   2) One fused pass reads x exactly once: fp32 WMMA (v_wmma_f32_16x16x4_f32,
//      full precision parity with the fp32 reference), softmax over E=32 via
//      wave32 half-shuffles, z-loss atomics, probs stored transposed [G,E,T].
//      W is staged in LDS in a pre-paired layout so each B fragment is a
//      single aligned ds_load_b64 into an even VGPR pair (no assembly movs).
//   3) A tiny top-k pass (256 blocks) scatters the 2 x 16384 nonzeros.

typedef float v2f __attribute__((ext_vector_type(2)));
typedef float v8f __attribute__((ext_vector_type(8)));

#define RG 8
#define RT 4096
#define RH 1024
#define RE 32
#define RC 64
#define RGT (RG * RT)
#define RGTEC ((long long)RG * RT * RE * RC)

// ---------------------------------------------------------------------------
// Pass 1: logits = x @ W + b (WMMA f32 16x16x4), softmax over E, z-loss.
// Block = 256 threads (8 waves). Each wave owns one 16-row M-tile of the
// [GT x E] logits; grid.x = GT/16/8 = 256. W (128 KB) staged in LDS as
// K-pairs: Wl2[k2 * RE + n] = { W[2*k2][n], W[2*k2+1][n] }.
// ---------------------------------------------------------------------------
__global__ __launch_bounds__(256) void router_gemm_softmax(
    const float* __restrict__ x,        // [G*T, H] row-major
    const float* __restrict__ W,        // [H, E] row-major
    const float* __restrict__ bias,     // [E]
    float* __restrict__ probs_et,       // ws: [G, E, T]
    float* __restrict__ zloss)          // scalar accumulator (pre-zeroed)
{
    extern __shared__ v2f Wl2[];        // (H/2)*E v2f = 128 KB (dynamic LDS)

    const int tid = threadIdx.x;
    // Stage W as K-pairs. Thread p -> (k2 = p/32, n = p%32): both global
    // reads are coalesced 128 B per 32-thread group; LDS writes are
    // consecutive 8 B slots (conflict-free).
    for (int p = tid; p < (RH / 2) * RE; p += 256) {
        const int k2 = p >> 5;
        const int n  = p & 31;
        v2f w;
        w.x = W[(2 * k2) * RE + n];
        w.y = W[(2 * k2 + 1) * RE + n];
        Wl2[p] = w;
    }
    __syncthreads();

    const int wave   = tid >> 5;
    const int lane   = tid & 31;
    const int hi16   = lane >> 4;       // 0: lanes 0-15, 1: lanes 16-31
    const int lane16 = lane & 15;

    const int tile = blockIdx.x * 8 + wave;       // M-tile over flattened GT
    const long long row0 = (long long)tile * 16;  // T%16==0 -> never crosses g
    const int g  = (int)(row0 / RT);
    const int t0 = (int)(row0 % RT);

    // A-fragment addressing (ISA 32-bit A 16x4 layout):
    //   lane holds row M = lane&15; lanes 0-15: K = kc,kc+1; lanes 16-31: kc+2,kc+3.
    const float* xrow = x + (row0 + lane16) * RH;
    const int koff = hi16 * 2;          // this lane's K sub-offset (0 or 2)

    v8f c0 = {};                        // experts 0..15
    v8f c1 = {};                        // experts 16..31

    for (int kc0 = 0; kc0 < RH; kc0 += 64) {
        // Emit global_prefetch_b8 for the next 256 B span of this lane's row.
        __builtin_prefetch(xrow + kc0 + 64, 0, 1);
        // Base of this chunk's B fragments for this lane; inner offsets are
        // compile-time immediates (256 B per K-pair, 8 KB span per chunk).
        const v2f* wb = &Wl2[((kc0 + koff) >> 1) * RE + lane16];
        #pragma unroll
        for (int kk = 0; kk < 64; kk += 4) {
            v2f a = *(const v2f*)(xrow + kc0 + kk + koff);   // 8 B load
            v2f b0 = wb[(kk >> 1) * RE];                     // ds_load_b64
            v2f b1 = wb[(kk >> 1) * RE + 16];                // ds_load_b64
            c0 = __builtin_amdgcn_wmma_f32_16x16x4_f32(
                     false, a, false, b0, (short)0, c0, false, false);
            c1 = __builtin_amdgcn_wmma_f32_16x16x4_f32(
                     false, a, false, b1, (short)0, c1, false, false);
        }
    }

    // C/D layout: element i of a lane = (row M = i + 8*hi16, expert = lane16
    // [+16 for c1]). Each logits row lives in element i across one 16-lane
    // half -> softmax reduces with xor-shuffles of width 16.
    const float be0 = bias[lane16];
    const float be1 = bias[lane16 + 16];
    float zacc = 0.0f;
    v8f p0, p1;
    #pragma unroll
    for (int i = 0; i < 8; ++i) {
        float l0 = c0[i] + be0;
        float l1 = c1[i] + be1;
        float m = fmaxf(l0, l1);
        m = fmaxf(m, __shfl_xor(m, 1, 16));
        m = fmaxf(m, __shfl_xor(m, 2, 16));
        m = fmaxf(m, __shfl_xor(m, 4, 16));
        m = fmaxf(m, __shfl_xor(m, 8, 16));
        // v_exp_f32 / v_log_f32 transcendentals (co-execute with VALU);
        // exp is monotone so top-k ordering is unchanged.
        float e0 = __expf(l0 - m);
        float e1 = __expf(l1 - m);
        float s = e0 + e1;
        s += __shfl_xor(s, 1, 16);
        s += __shfl_xor(s, 2, 16);
        s += __shfl_xor(s, 4, 16);
        s += __shfl_xor(s, 8, 16);
        const float inv = 1.0f / s;
        p0[i] = e0 * inv;
        p1[i] = e1 * inv;
        const float lz = m + __logf(s); // logsumexp of this row
        zacc += lz * lz;
    }
    if (lane16 == 0) {                  // one lane per half: 8 rows each
        atomicAdd(zloss, zacc * (1.0f / (float)RGT));
    }

    // Store probs transposed [G, E, T]: element i maps to t = t0 + 8*hi16 + i,
    // contiguous per lane -> one 32 B store per fragment.
    float* dst0 = probs_et + ((long long)g * RE + lane16) * RT + t0 + hi16 * 8;
    float* dst1 = probs_et + ((long long)g * RE + lane16 + 16) * RT + t0 + hi16 * 8;
    *(v8f*)dst0 = p0;
    *(v8f*)dst1 = p1;
}

// ---------------------------------------------------------------------------
// Pass 2: per (g,e), select top-C tokens (descending, ties -> smaller index,
// matching lax.top_k) and scatter gate / 1.0 into combine / dispatch.
// One block per (g,e): 256 blocks x 256 threads. Probs row cached in LDS.
// ---------------------------------------------------------------------------
__global__ __launch_bounds__(256) void expert_topk_scatter(
    const float* __restrict__ probs_et, // [G, E, T]
    float* __restrict__ combine,        // [G, T, E, C] (pre-zeroed)
    float* __restrict__ dispatch)       // [G, T, E, C] (pre-zeroed)
{
    __shared__ float vals[RT];          // 16 KB
    __shared__ float rv[256];
    __shared__ int   ri[256];

    const int ge = blockIdx.x;
    const int g = ge / RE;
    const int e = ge % RE;
    const int tid = threadIdx.x;

    const float* src = probs_et + (long long)ge * RT;
    for (int i = tid; i < RT; i += 256) vals[i] = src[i];
    __syncthreads();

    for (int c = 0; c < RC; ++c) {
        // Local argmax over strided slice; scanning ascending j with a strict
        // '>' keeps the smallest index on ties.
        float best = -1.0f;
        int   bidx = RT;
        for (int j = tid; j < RT; j += 256) {
            const float v = vals[j];
            if (v > best) { best = v; bidx = j; }
        }
        rv[tid] = best;
        ri[tid] = bidx;
        __syncthreads();
        for (int s = 128; s > 0; s >>= 1) {
            if (tid < s) {
                const float ov = rv[tid + s];
                const int   oi = ri[tid + s];
                if (ov > rv[tid] || (ov == rv[tid] && oi < ri[tid])) {
                    rv[tid] = ov;
                    ri[tid] = oi;
                }
            }
            __syncthreads();
        }
        if (tid == 0) {
            const int   t    = ri[0];
            const float gate = rv[0];
            const long long off =
                (((long long)g * RT + t) * RE + e) * RC + c;
            combine[off]  = gate;       // dispatch * gate at the hot entry
            dispatch[off] = 1.0f;
            vals[t] = -1.0f;            // remove from candidate set
        }
        __syncthreads();
    }
}

extern "C" void kernel_launch(void* const* d_in, const int* in_sizes, int n_in,
                              void* d_out, int out_size, void* d_ws, size_t ws_size,
                              hipStream_t stream) {
    (void)in_sizes; (void)n_in; (void)ws_size;

    const float* x    = (const float*)d_in[0];   // [G, T, H] fp32
    const float* W    = (const float*)d_in[1];   // [H, E]   fp32
    const float* bias = (const float*)d_in[2];   // [E]      fp32
    // d_in[3] = num_experts, d_in[4] = expert_capacity (compile-time here)

    float* out      = (float*)d_out;
    float* combine  = out;                       // [G,T,E,C]
    float* dispatch = out + RGTEC;               // [G,T,E,C]
    float* zloss    = out + 2 * RGTEC;           // scalar
    float* probs    = (float*)d_ws;              // [G,E,T] = 4 MB scratch

    // Zero the (sparse) outputs and the z-loss accumulator: pure store BW.
    hipMemsetAsync(d_out, 0, (size_t)out_size * sizeof(float), stream);

    // Pass 1: 256 blocks x 256 threads, 128 KB dynamic LDS for W (pre-paired).
    router_gemm_softmax<<<RGT / (16 * 8), 256, (RH / 2) * RE * sizeof(v2f), stream>>>(
        x, W, bias, probs, zloss);

    // Pass 2: one block per (g, e).
    expert_topk_scatter<<<RG * RE, 256, 0, stream>>>(probs, combine, dispatch);
}